// ImageNormalization3D_9156870275417
// MI455X (gfx1250) — compile-verified
//
#include <hip/hip_runtime.h>
#include <math.h>

// ImageNormalization3D for x:(4,7,768,768,2) f32 — memory-bound separable box
// filter. Min HBM traffic ~264MB => ~11.3us at 23.3TB/s; intermediates are
// 66-72MB scalar fields (394 MiB workspace), L2-competitive (192MB L2).
// W-direction 61-box runs on the matrix pipe: V_WMMA_F32_16X16X4_F32 with a
// banded ones B-matrix. A-fields are zero-padded in W, each wave computes TWO
// adjacent 16x16 output tiles from one shared 23-step k-window (fully
// unrolled, base-pointer + immediate-offset loads, branch-free).

#define NN 4
#define DD 7
#define HH 768
#define WW 768
#define HWs (HH * WW)        // 589824, unpadded d-stride
#define RAD 30               // 61-wide box radius
#define CH 64                // H-chunk per thread in running-window kernels
#define PAD 32               // zero pad columns each side (even, >= RAD)
#define WP (WW + 2 * PAD)    // 832, padded row stride
#define HWP (HH * WP)        // padded d-stride
#define RTOT (NN * DD * HH)  // 21504 flattened rows

static_assert(HH % CH == 0, "chunking");
static_assert(CH > RAD, "warmup writes must stay in chunk");

constexpr float INVK = 1.0f / (7.0f * 61.0f * 61.0f * 2.0f); // 1/52094
constexpr float EPSF = 1e-7f;

typedef __attribute__((ext_vector_type(2))) float v2f;
typedef __attribute__((ext_vector_type(8))) float v8f;

#define WMMA_F32_4(A, B, C) \
  __builtin_amdgcn_wmma_f32_16x16x4_f32(false, (A), false, (B), (short)0, (C), false, false)

// band mask pair for distance e = col - k (compile-time t folds constants)
__device__ __forceinline__ v2f band_b(int e) {
  v2f b;
  b.x = ((unsigned)(e + RAD) < 61u) ? 1.f : 0.f;
  b.y = ((unsigned)(e - 1 + RAD) < 61u) ? 1.f : 0.f;
  return b;
}

// depth box: window 7 (pl=pr=3, zero pad == clipped window), D=7
__device__ __forceinline__ void dbox7(const float* s, float* b) {
  float ps[8];
  ps[0] = 0.f;
#pragma unroll
  for (int d = 0; d < 7; ++d) ps[d + 1] = ps[d] + s[d];
#pragma unroll
  for (int d = 0; d < 7; ++d) {
    const int hi = (d + 3 > 6) ? 7 : (d + 4);
    const int lo = (d - 3 < 0) ? 0 : (d - 3);
    b[d] = ps[hi] - ps[lo];
  }
}

// ---------------------------------------------------------------------------
// K0: zero the 32-column left/right pad margins of the three padded fields.
// Pads are never written by K1/K3, so zeroing once per launch covers both the
// A-phase (A1,A2) and B-phase (B1,B2,B3 reuse the same slots).
// ---------------------------------------------------------------------------
__global__ void __launch_bounds__(256)
k0_zero_pads(float* __restrict__ f0, float* __restrict__ f1,
             float* __restrict__ f2) {
  const int i = blockIdx.x * 256 + threadIdx.x;  // RTOT*64 per field
  const int row = i >> 6;
  const int c = i & 63;
  const int col = (c < PAD) ? c : (c + WW);      // [0,32) U [800,832)
  const size_t o = (size_t)row * WP + col;
  float* f = (blockIdx.y == 0) ? f0 : (blockIdx.y == 1) ? f1 : f2;
  f[o] = 0.f;
}

// ---------------------------------------------------------------------------
// K1: read x, produce S1 = sum_c x (raw, unpadded), A1 = Hbox(Dbox(S1)) and
// A2 = Hbox(Dbox(S2)) (padded layout). One thread per (n, w, h-chunk);
// running 61-window along H, coalesced over w; trailing re-reads hit L2.
// ---------------------------------------------------------------------------
__global__ void __launch_bounds__(64)
k1_chsum_dbox_hbox(const float* __restrict__ x, float* __restrict__ S1,
                   float* __restrict__ A1, float* __restrict__ A2) {
  const float2* x2 = (const float2*)x;           // [n][d][h][w] of float2 (c=0,1)
  const int w  = blockIdx.x * 64 + threadIdx.x;
  const int h0 = blockIdx.y * CH;
  const int n  = blockIdx.z;
  const size_t fbase = (size_t)n * DD * HWs + w;          // unpadded (n,0,0,w)
  const size_t pbase = (size_t)n * DD * HWP + PAD + w;    // padded   (n,0,0,w)

  float win1[7], win2[7];
#pragma unroll
  for (int d = 0; d < 7; ++d) { win1[d] = 0.f; win2[d] = 0.f; }

  float s1[7], s2[7], b1[7], b2[7];

  // warmup rows [h0-30, h0+29]
  for (int hr = h0 - RAD; hr < h0 + RAD; ++hr) {
    if (hr < 0) continue;                        // upper bound always < HH here
    const size_t rb = fbase + (size_t)hr * WW;
#pragma unroll
    for (int d = 0; d < 7; ++d) {
      const float2 v = x2[rb + (size_t)d * HWs];
      s1[d] = v.x + v.y;
      s2[d] = v.x * v.x + v.y * v.y;
    }
    dbox7(s1, b1); dbox7(s2, b2);
#pragma unroll
    for (int d = 0; d < 7; ++d) { win1[d] += b1[d]; win2[d] += b2[d]; }
    if (hr >= h0) {
#pragma unroll
      for (int d = 0; d < 7; ++d) S1[rb + (size_t)d * HWs] = s1[d];
    }
  }

  for (int h = h0; h < h0 + CH; ++h) {
    const int hl = h + RAD;                      // leading row
    if (hl < HH) {
      const size_t rb = fbase + (size_t)hl * WW;
      if (hl + 1 < HH) __builtin_prefetch(&x2[rb + WW], 0, 1); // global_prefetch_b8
#pragma unroll
      for (int d = 0; d < 7; ++d) {
        const float2 v = x2[rb + (size_t)d * HWs];
        s1[d] = v.x + v.y;
        s2[d] = v.x * v.x + v.y * v.y;
      }
      dbox7(s1, b1); dbox7(s2, b2);
#pragma unroll
      for (int d = 0; d < 7; ++d) { win1[d] += b1[d]; win2[d] += b2[d]; }
      if (hl < h0 + CH) {
#pragma unroll
        for (int d = 0; d < 7; ++d) S1[rb + (size_t)d * HWs] = s1[d];
      }
    }
    {
      const size_t ob = pbase + (size_t)h * WP;
#pragma unroll
      for (int d = 0; d < 7; ++d) {
        A1[ob + (size_t)d * HWP] = win1[d];
        A2[ob + (size_t)d * HWP] = win2[d];
      }
    }
    const int ht = h - RAD;                      // trailing row leaves window
    if (ht >= 0) {
      const size_t rb = fbase + (size_t)ht * WW;
#pragma unroll
      for (int d = 0; d < 7; ++d) {
        const float2 v = x2[rb + (size_t)d * HWs];
        s1[d] = v.x + v.y;
        s2[d] = v.x * v.x + v.y * v.y;
      }
      dbox7(s1, b1); dbox7(s2, b2);
#pragma unroll
      for (int d = 0; d < 7; ++d) { win1[d] -= b1[d]; win2[d] -= b2[d]; }
    }
  }
}

// ---------------------------------------------------------------------------
// K2: W-direction 61-box of padded A1, A2 via WMMA banded matmul. One wave
// per 16x32 output block (two adjacent 16x16 tiles sharing one 92-wide
// k-window = 23 aligned k-steps, fully unrolled). A f32 16x4: lanes 0-15 =
// rows, VGPR v = k_local v + 2*(lane>=16); B 4x16 (C/D analogy): VGPR v = row
// v + 2*(lane>=16), col = lane&15. Writes m, q2 = boxsum*INVK (unpadded).
// Steps t<=18 feed tile0, t>=4 feed tile1 (compile-time split).
// ---------------------------------------------------------------------------
__global__ void __launch_bounds__(128)
k2_wbox_wmma2(const float* __restrict__ A1, const float* __restrict__ A2,
              float* __restrict__ m, float* __restrict__ q2) {
  const int lane = threadIdx.x;
  const int half = lane >> 4;
  const int n16  = lane & 15;
  const int r0 = (blockIdx.y * 4 + threadIdx.y) * 16;
  const int j0 = blockIdx.x * 32;
  const int rowA = r0 + n16;
  const float* pa1 = A1 + (size_t)rowA * WP + (PAD - RAD + 2 * half + j0);
  const float* pa2 = A2 + (size_t)rowA * WP + (PAD - RAD + 2 * half + j0);
  const int u = n16 + RAD - 2 * half;            // d0(t) = u - 4t (tile0)

  v8f c10 = {0.f,0.f,0.f,0.f,0.f,0.f,0.f,0.f};   // field1, tile0
  v8f c20 = {0.f,0.f,0.f,0.f,0.f,0.f,0.f,0.f};   // field2, tile0
  v8f c11 = {0.f,0.f,0.f,0.f,0.f,0.f,0.f,0.f};   // field1, tile1
  v8f c21 = {0.f,0.f,0.f,0.f,0.f,0.f,0.f,0.f};   // field2, tile1

#pragma unroll
  for (int t = 0; t < 23; ++t) {                 // k in [j0-30, j0+61]
    const v2f a1 = *reinterpret_cast<const v2f*>(pa1 + 4 * t);
    const v2f a2 = *reinterpret_cast<const v2f*>(pa2 + 4 * t);
    if (t <= 18) {                               // tile0 band reaches j0+45
      const v2f b = band_b(u - 4 * t);
      c10 = WMMA_F32_4(a1, b, c10);
      c20 = WMMA_F32_4(a2, b, c20);
    }
    if (t >= 4) {                                // tile1 band starts at j0-14
      const v2f b = band_b(u + 16 - 4 * t);
      c11 = WMMA_F32_4(a1, b, c11);
      c21 = WMMA_F32_4(a2, b, c21);
    }
  }
#pragma unroll
  for (int v = 0; v < 8; ++v) {
    const int row = r0 + v + 8 * half;           // C/D layout
    const size_t o = (size_t)row * WW + (j0 + n16);
    m[o]       = c10[v] * INVK;
    q2[o]      = c20[v] * INVK;
    m[o + 16]  = c11[v] * INVK;
    q2[o + 16] = c21[v] * INVK;
  }
}

// ---------------------------------------------------------------------------
// K3: from m and raw S1 (unpadded) produce B1 = Hbox(Dbox(m)),
// B2 = Hbox(Dbox(m*S1)), B3 = Hbox(Dbox(m*m)) in padded layout.
// ---------------------------------------------------------------------------
__global__ void __launch_bounds__(64)
k3_prep_dbox_hbox(const float* __restrict__ m, const float* __restrict__ S1,
                  float* __restrict__ B1, float* __restrict__ B2,
                  float* __restrict__ B3) {
  const int w  = blockIdx.x * 64 + threadIdx.x;
  const int h0 = blockIdx.y * CH;
  const int n  = blockIdx.z;
  const size_t fbase = (size_t)n * DD * HWs + w;
  const size_t pbase = (size_t)n * DD * HWP + PAD + w;

  float w1[7], w2[7], w3[7];
#pragma unroll
  for (int d = 0; d < 7; ++d) { w1[d] = 0.f; w2[d] = 0.f; w3[d] = 0.f; }

  float t1[7], t2[7], t3[7], b1[7], b2[7], b3[7];

  auto readrow = [&](int hr) {
    const size_t rb = fbase + (size_t)hr * WW;
#pragma unroll
    for (int d = 0; d < 7; ++d) {
      const float mv = m[rb + (size_t)d * HWs];
      const float sv = S1[rb + (size_t)d * HWs];
      t1[d] = mv;
      t2[d] = mv * sv;
      t3[d] = mv * mv;
    }
    dbox7(t1, b1); dbox7(t2, b2); dbox7(t3, b3);
  };

  for (int hr = h0 - RAD; hr < h0 + RAD; ++hr) {
    if (hr < 0) continue;
    readrow(hr);
#pragma unroll
    for (int d = 0; d < 7; ++d) { w1[d] += b1[d]; w2[d] += b2[d]; w3[d] += b3[d]; }
  }
  for (int h = h0; h < h0 + CH; ++h) {
    const int hl = h + RAD;
    if (hl < HH) {
      if (hl + 1 < HH) __builtin_prefetch(&m[fbase + (size_t)(hl + 1) * WW], 0, 1);
      readrow(hl);
#pragma unroll
      for (int d = 0; d < 7; ++d) { w1[d] += b1[d]; w2[d] += b2[d]; w3[d] += b3[d]; }
    }
    {
      const size_t ob = pbase + (size_t)h * WP;
#pragma unroll
      for (int d = 0; d < 7; ++d) {
        B1[ob + (size_t)d * HWP] = w1[d];
        B2[ob + (size_t)d * HWP] = w2[d];
        B3[ob + (size_t)d * HWP] = w3[d];
      }
    }
    const int ht = h - RAD;
    if (ht >= 0) {
      readrow(ht);
#pragma unroll
      for (int d = 0; d < 7; ++d) { w1[d] -= b1[d]; w2[d] -= b2[d]; w3[d] -= b3[d]; }
    }
  }
}

// ---------------------------------------------------------------------------
// K4: W-box of padded B1,B2,B3 via WMMA (two tiles per wave), fused stats:
//   c1 = m - (2/K)*F(m);  c2 = q2 - (2/K)*(F(m*S1) - F(m*m))
//   R  = 1 / (sqrt(c2 - c1^2) + 1e-7)
// ---------------------------------------------------------------------------
__global__ void __launch_bounds__(128)
k4_wbox_wmma3_stats(const float* __restrict__ B1, const float* __restrict__ B2,
                    const float* __restrict__ B3, const float* __restrict__ m,
                    const float* __restrict__ q2, float* __restrict__ R) {
  const int lane = threadIdx.x;
  const int half = lane >> 4;
  const int n16  = lane & 15;
  const int r0 = (blockIdx.y * 4 + threadIdx.y) * 16;
  const int j0 = blockIdx.x * 32;
  const int rowA = r0 + n16;
  const float* p1 = B1 + (size_t)rowA * WP + (PAD - RAD + 2 * half + j0);
  const float* p2 = B2 + (size_t)rowA * WP + (PAD - RAD + 2 * half + j0);
  const float* p3 = B3 + (size_t)rowA * WP + (PAD - RAD + 2 * half + j0);
  const int u = n16 + RAD - 2 * half;

  v8f cm0 = {0.f,0.f,0.f,0.f,0.f,0.f,0.f,0.f};
  v8f cs0 = {0.f,0.f,0.f,0.f,0.f,0.f,0.f,0.f};
  v8f cq0 = {0.f,0.f,0.f,0.f,0.f,0.f,0.f,0.f};
  v8f cm1 = {0.f,0.f,0.f,0.f,0.f,0.f,0.f,0.f};
  v8f cs1 = {0.f,0.f,0.f,0.f,0.f,0.f,0.f,0.f};
  v8f cq1 = {0.f,0.f,0.f,0.f,0.f,0.f,0.f,0.f};

#pragma unroll
  for (int t = 0; t < 23; ++t) {
    const v2f a1 = *reinterpret_cast<const v2f*>(p1 + 4 * t);
    const v2f a2 = *reinterpret_cast<const v2f*>(p2 + 4 * t);
    const v2f a3 = *reinterpret_cast<const v2f*>(p3 + 4 * t);
    if (t <= 18) {
      const v2f b = band_b(u - 4 * t);
      cm0 = WMMA_F32_4(a1, b, cm0);
      cs0 = WMMA_F32_4(a2, b, cs0);
      cq0 = WMMA_F32_4(a3, b, cq0);
    }
    if (t >= 4) {
      const v2f b = band_b(u + 16 - 4 * t);
      cm1 = WMMA_F32_4(a1, b, cm1);
      cs1 = WMMA_F32_4(a2, b, cs1);
      cq1 = WMMA_F32_4(a3, b, cq1);
    }
  }
#pragma unroll
  for (int v = 0; v < 8; ++v) {
    const int row = r0 + v + 8 * half;
    const size_t o = (size_t)row * WW + (j0 + n16);
    {
      const float mv  = m[o];
      const float qv  = q2[o];
      const float c1v = mv - 2.0f * INVK * cm0[v];
      const float c2v = qv - 2.0f * INVK * (cs0[v] - cq0[v]);
      const float var = c2v - c1v * c1v;
      R[o] = 1.0f / (sqrtf(var) + EPSF);
    }
    {
      const float mv  = m[o + 16];
      const float qv  = q2[o + 16];
      const float c1v = mv - 2.0f * INVK * cm1[v];
      const float c2v = qv - 2.0f * INVK * (cs1[v] - cq1[v]);
      const float var = c2v - c1v * c1v;
      R[o + 16] = 1.0f / (sqrtf(var) + EPSF);
    }
  }
}

// ---------------------------------------------------------------------------
// K5: out[n,d,h,w,c] = (x - m) * R, vectorized over channel pairs.
// ---------------------------------------------------------------------------
__global__ void __launch_bounds__(256)
k5_finalize(const float* __restrict__ x, const float* __restrict__ m,
            const float* __restrict__ R, float* __restrict__ out) {
  const size_t i = (size_t)blockIdx.x * 256 + threadIdx.x;   // F = 64512*256 exactly
  const float2 xv = ((const float2*)x)[i];
  const float mv = m[i];
  const float rv = R[i];
  float2 o;
  o.x = (xv.x - mv) * rv;
  o.y = (xv.y - mv) * rv;
  ((float2*)out)[i] = o;
}

extern "C" void kernel_launch(void* const* d_in, const int* in_sizes, int n_in,
                              void* d_out, int out_size, void* d_ws, size_t ws_size,
                              hipStream_t stream) {
  (void)in_sizes; (void)n_in; (void)out_size; (void)ws_size;
  const float* x = (const float*)d_in[0];
  float* out = (float*)d_out;
  float* ws = (float*)d_ws;

  const size_t F  = (size_t)NN * DD * HH * WW;  // 16,515,072 unpadded field
  const size_t FP = (size_t)RTOT * WP;          // 17,891,328 padded field
  // Workspace: 3 unpadded + 3 padded fields = ~394 MiB, with slot reuse:
  float* S1  = ws;                 // K1 out (raw) -> K3 in; reused as R (K4 out, K5 in)
  float* A1p = ws + F;             // padded: K1 out -> K2 in; reused as B1 (K3->K4)
  float* A2p = ws + F + FP;        // padded: K1 out -> K2 in; reused as B2
  float* mf  = ws + F + 2 * FP;    // K2 out (mean field), live to K5
  float* q2  = mf + F;             // K2 out (F(S2)/K), K4 in
  float* B3p = q2 + F;             // padded: K3 out -> K4 in

  const dim3 gZ((RTOT * 64) / 256, 3);          // (5376, 3) pad-zeroing
  const dim3 gHB(WW / 64, HH / CH, NN);         // (12, 12, 4)
  const dim3 bHB(64);
  const dim3 gW(WW / 32, RTOT / (16 * 4));      // (24, 336): 16x32 per wave
  const dim3 bW(32, 4);                         // one wave32 per 16x32 block

  k0_zero_pads<<<gZ, 256, 0, stream>>>(A1p, A2p, B3p);
  k1_chsum_dbox_hbox<<<gHB, bHB, 0, stream>>>(x, S1, A1p, A2p);
  k2_wbox_wmma2<<<gW, bW, 0, stream>>>(A1p, A2p, mf, q2);
  k3_prep_dbox_hbox<<<gHB, bHB, 0, stream>>>(mf, S1, A1p, A2p, B3p);
  k4_wbox_wmma3_stats<<<gW, bW, 0, stream>>>(A1p, A2p, B3p, mf, q2, /*R=*/S1);
  k5_finalize<<<(unsigned)(F / 256), 256, 0, stream>>>(x, mf, S1, out);
}